// HCHead_36112085025014
// MI455X (gfx1250) — compile-verified
//
#include <hip/hip_runtime.h>
#include <stdint.h>

// MI455X / gfx1250. Bandwidth-bound fused RMS-norm + dynamic 4-way stream mix.
// Traffic floor: 512MB x-read + 128MB y-write ~= 27us @ 23.3 TB/s.
// Strategy: 8 tokens/block, bf16 stash of xf in 320KB CDNA5 LDS (read x ONCE),
// logits matmul via V_WMMA_F32_16X16X4_F32 (f32 -> exact reference numerics).
// Unused WMMA D rows(8-15)/cols(4-15) are garbage-by-design: never read, so no
// lane masking needed in the hot loop (addresses clamped in-bounds only).

typedef float    v2f __attribute__((ext_vector_type(2)));
typedef float    v4f __attribute__((ext_vector_type(4)));
typedef float    v8f __attribute__((ext_vector_type(8)));
typedef uint32_t v2u __attribute__((ext_vector_type(2)));

#define MULT        4
#define HIDDEN      4096
#define C_DIM       16384           // MULT*HIDDEN
#define TOK_PER_BLK 8
#define NTHREADS    512
#define NWAVES      16              // 512 / wave32
#define ROW_WORDS   8194            // 16384 bf16 / 2 per dword, +2 dword pad (bank spread)

__device__ __forceinline__ uint32_t bf16_rne(float v) {
    uint32_t u = __float_as_uint(v);
    u += 0x7FFFu + ((u >> 16) & 1u);   // round-to-nearest-even (inputs are finite)
    return u >> 16;
}
// Pack two f32 -> two bf16 (RNE) in one dword; low half = first arg.
__device__ __forceinline__ uint32_t pack2_bf16(float a, float b) {
#if __has_builtin(__builtin_amdgcn_cvt_pk_bf16_f32)
    typedef __bf16 v2bf __attribute__((ext_vector_type(2)));
    v2bf p = __builtin_amdgcn_cvt_pk_bf16_f32(a, b);   // S0 -> D[15:0], S1 -> D[31:16]
    uint32_t u;
    __builtin_memcpy(&u, &p, 4);
    return u;
#else
    return bf16_rne(a) | (bf16_rne(b) << 16);
#endif
}
__device__ __forceinline__ float bf_lo(uint32_t w) { return __uint_as_float(w << 16); }
__device__ __forceinline__ float bf_hi(uint32_t w) { return __uint_as_float(w & 0xFFFF0000u); }

__global__ __launch_bounds__(NTHREADS)
void hchead_kernel(const float* __restrict__ x,
                   const float* __restrict__ fn,
                   const float* __restrict__ scale,
                   const float* __restrict__ base,
                   float* __restrict__ out)
{
    __shared__ uint32_t stash[TOK_PER_BLK][ROW_WORDS];     // bf16 xf, 2 per dword
    __shared__ float    SP[NWAVES][TOK_PER_BLK];           // sumsq partials per wave
    __shared__ float    P[NWAVES][TOK_PER_BLK][MULT];      // dot partials per wave
    __shared__ float    wbuf[TOK_PER_BLK][MULT];           // final mixing weights

    const int tid  = threadIdx.x;
    const int wave = tid >> 5;
    const int lane = tid & 31;
    const int tok0 = blockIdx.x * TOK_PER_BLK;

    // ---------------- pass 0: stream x once (NT), bf16-cast into LDS, sumsq ----
    float s[TOK_PER_BLK];
    #pragma unroll
    for (int r = 0; r < TOK_PER_BLK; ++r) s[r] = 0.0f;

    #pragma unroll
    for (int r = 0; r < TOK_PER_BLK; ++r) {
        const v4f* xr = (const v4f*)(x + (size_t)(tok0 + r) * C_DIM);
        #pragma unroll
        for (int k = 0; k < 8; ++k) {
            const int idx = tid * 4 + k * 2048;            // element index (mult of 4)
            v4f v = __builtin_nontemporal_load(xr + (idx >> 2));
            const uint32_t w0 = pack2_bf16(v.x, v.y);
            const uint32_t w1 = pack2_bf16(v.z, v.w);
            const float f0 = bf_lo(w0), f1 = bf_hi(w0);
            const float f2 = bf_lo(w1), f3 = bf_hi(w1);
            s[r] = fmaf(f0, f0, s[r]); s[r] = fmaf(f1, f1, s[r]);
            s[r] = fmaf(f2, f2, s[r]); s[r] = fmaf(f3, f3, s[r]);
            v2u pw = { w0, w1 };
            *(v2u*)&stash[r][idx >> 1] = pw;               // ds_store_b64
        }
    }
    #pragma unroll
    for (int r = 0; r < TOK_PER_BLK; ++r) {
        float v = s[r];
        #pragma unroll
        for (int m = 16; m > 0; m >>= 1) v += __shfl_xor(v, m, 32);
        if (lane == 0) SP[wave][r] = v;
    }
    __syncthreads();

    // ---------------- pass 1: logits via f32 WMMA  D[r][m] += X[r][k]*fn[m][k] ----
    // A (16x4 f32): lanes 0-15 hold K=0,1 ; lanes 16-31 hold K=2,3 (rows = tokens)
    // B (4x16 f32): VGPR0 = B[K=2h][col], VGPR1 = B[K=2h+1][col]   (cols = m)
    // No value masking: D rows >=8 and cols >=4 are never read (garbage OK, finite).
    const int  tokRow = lane & 15;
    const int  half   = lane >> 4;
    const int  tokC   = (tokRow < TOK_PER_BLK) ? tokRow : (TOK_PER_BLK - 1); // in-bounds clamp
    const int  jC     = (tokRow < MULT) ? tokRow : (MULT - 1);               // in-bounds clamp
    const float* fnRow = fn + (size_t)jC * C_DIM;

    v8f acc0 = {0.f,0.f,0.f,0.f,0.f,0.f,0.f,0.f};
    v8f acc1 = acc0;

    const int cBase = wave * 256;                          // 256 K-chunks (of 4) per wave
    for (int it = 0; it < 128; ++it) {
        const int c0 = cBase + 2 * it;
        {
            const uint32_t aw = stash[tokC][2 * c0 + half];
            v2f a; a.x = bf_lo(aw); a.y = bf_hi(aw);
            const v2f b = *(const v2f*)(fnRow + 4 * c0 + 2 * half);
            acc0 = __builtin_amdgcn_wmma_f32_16x16x4_f32(false, a, false, b,
                                                         (short)0, acc0, false, false);
        }
        {
            const int c1 = c0 + 1;
            const uint32_t aw = stash[tokC][2 * c1 + half];
            v2f a; a.x = bf_lo(aw); a.y = bf_hi(aw);
            const v2f b = *(const v2f*)(fnRow + 4 * c1 + 2 * half);
            acc1 = __builtin_amdgcn_wmma_f32_16x16x4_f32(false, a, false, b,
                                                         (short)0, acc1, false, false);
        }
    }
    v8f acc = acc0 + acc1;
    // D rows 0-7 live in VGPRs 0-7 at lanes 0-15 (cols); we need cols 0-3 only.
    if (lane < MULT) {
        #pragma unroll
        for (int r = 0; r < TOK_PER_BLK; ++r) P[wave][r][lane] = acc[r];
    }
    __syncthreads();

    // ---------------- tiny per-token math: rms -> logits -> mixing weights ------
    if (tid < TOK_PER_BLK) {
        const int r = tid;
        float st = 0.0f;
        #pragma unroll
        for (int w = 0; w < NWAVES; ++w) st += SP[w][r];
        const float rms = sqrtf(st * (1.0f / (float)C_DIM) + 1e-6f);
        const float sc  = scale[0];
        float a[MULT]; float den = 1e-6f;
        #pragma unroll
        for (int m = 0; m < MULT; ++m) {
            float d = 0.0f;
            #pragma unroll
            for (int w = 0; w < NWAVES; ++w) d += P[w][r][m];
            a[m] = fmaf(sc, d / rms, base[m]);
            den += fabsf(a[m]);
        }
        #pragma unroll
        for (int m = 0; m < MULT; ++m) wbuf[r][m] = a[m] / den;
    }
    __syncthreads();

    // ---------------- pass 2: y[h] = sum_m w[m] * xf[m*4096 + h]  (from LDS) ----
    #pragma unroll
    for (int r = 0; r < TOK_PER_BLK; ++r) {
        const float w0 = wbuf[r][0], w1 = wbuf[r][1];
        const float w2 = wbuf[r][2], w3 = wbuf[r][3];
        v4f* yr = (v4f*)(out + (size_t)(tok0 + r) * HIDDEN);
        #pragma unroll
        for (int k = 0; k < 2; ++k) {
            const int h0 = tid * 4 + k * 2048;             // output h (mult of 4)
            v4f y = {0.f, 0.f, 0.f, 0.f};
            #pragma unroll
            for (int m = 0; m < MULT; ++m) {
                const int word = m * 2048 + (h0 >> 1);
                const uint32_t u0 = stash[r][word];
                const uint32_t u1 = stash[r][word + 1];
                const float wm = (m == 0) ? w0 : (m == 1) ? w1 : (m == 2) ? w2 : w3;
                y.x = fmaf(wm, bf_lo(u0), y.x);
                y.y = fmaf(wm, bf_hi(u0), y.y);
                y.z = fmaf(wm, bf_lo(u1), y.z);
                y.w = fmaf(wm, bf_hi(u1), y.w);
            }
            __builtin_nontemporal_store(y, yr + (h0 >> 2));
        }
    }
}

extern "C" void kernel_launch(void* const* d_in, const int* in_sizes, int n_in,
                              void* d_out, int out_size, void* d_ws, size_t ws_size,
                              hipStream_t stream) {
    const float* x     = (const float*)d_in[0];   // [T, 16384] f32
    const float* fn    = (const float*)d_in[1];   // [4, 16384] f32
    const float* scale = (const float*)d_in[2];   // [1] f32
    const float* base  = (const float*)d_in[3];   // [4] f32
    float* out = (float*)d_out;                   // [T, 4096] f32

    const int T = in_sizes[0] / C_DIM;            // 8192
    dim3 grid(T / TOK_PER_BLK);                   // 1024 blocks
    hchead_kernel<<<grid, NTHREADS, 0, stream>>>(x, fn, scale, base, out);
}